// UncertaintyEstimatorOCC_34600256537500
// MI455X (gfx1250) — compile-verified
//
#include <hip/hip_runtime.h>

// UncertaintyEstimatorOCC: per pred box, count in how many of 16 MC-dropout
// runs it has any IoU>0.5 match among that run's 2048 boxes; output = count/16.
//
// VALU-bound min/max workload (no matmul structure -> no WMMA). CDNA5 paths:
//  * TDM tensor_load_to_lds: one DMA op stages the strided 2D box tile
//    (2048 rows x 16B out of 24B-strided rows) into LDS  [TENSORcnt]
//  * global_load_async_to_lds_b64: coalesces each thread's pred row  [ASYNCcnt]
//  * wave32 ballot early-exit, LDS broadcast reads.

#define IOU_THR 0.5f
#define EPS_F   1e-7f
#define N_PRED  2048
#define T_RUNS  16
#define M_BOX   2048
#define BLOCK   128
#define TILES   (N_PRED / BLOCK)          // 16 pred tiles

typedef unsigned uint32x4 __attribute__((ext_vector_type(4)));
typedef unsigned uint32x8 __attribute__((ext_vector_type(8)));

__device__ __forceinline__ unsigned lds_addr_of(const void* p) {
    // Generic pointers to __shared__ carry the workgroup-relative LDS byte
    // offset in their low 32 bits (flat shared aperture layout).
    return (unsigned)(unsigned long long)p;
}

__global__ __launch_bounds__(BLOCK) void iou_match_kernel(
    const float* __restrict__ pred,           // (N_PRED, 6)
    const float* __restrict__ dropout_preds,  // (T_RUNS, M_BOX, 6)
    unsigned*    __restrict__ counts)         // (N_PRED)
{
    __shared__ float4 sbox[M_BOX];            // 32 KB: packed x1,y1,x2,y2
    __shared__ float  spred[BLOCK * 6];       // 3 KB: this tile's pred rows

    const int tid = threadIdx.x;
    const int i   = blockIdx.x * BLOCK + tid; // pred box index
    const int t   = blockIdx.y;               // dropout run index

    // ---- TDM: DMA the whole 2048x4f32 tile (from 6f32-strided rows) to LDS.
    // D# per CDNA5 ISA ch.8; 2D tensor -> groups 0,1 only. Issued by wave 0
    // (scalar branch via readfirstlane; tensor ops ignore EXEC).
    const unsigned long long ga =
        (unsigned long long)(dropout_preds + (size_t)t * M_BOX * 6);
    if (__builtin_amdgcn_readfirstlane((int)tid) == 0) {
        const unsigned lbase = lds_addr_of(&sbox[0]);
        uint32x4 g0 = {
            1u,                                              // count=1, user D#
            lbase,                                           // lds_addr
            (unsigned)ga,                                    // global_addr[31:0]
            (unsigned)((ga >> 32) & 0x01FFFFFFull) | (2u << 30) // ga[56:32] | type=2
        };
        uint32x8 g1 = {
            2u << 16,                    // data_size=2 (4B); wg_mask=0, no flags
            6u << 16,                    // tensor_dim0 = 6 elements (row length)
            (unsigned)M_BOX << 16,       // tensor_dim1 = 2048 (low16); dim0 hi=0
            4u << 16,                    // tile_dim0 = 4 elements; dim1 hi16=0
            (unsigned)M_BOX,             // tile_dim1 = 2048; tile_dim2=0
            6u,                          // tensor_dim0_stride = 6 elements
            0u,                          // stride0 hi16 = 0; stride1 lo16 = 0
            0u                           // stride1 hi = 0 (unused, 2D tile)
        };
        asm volatile("tensor_load_to_lds %0, %1" :: "s"(g0), "s"(g1) : "memory");
    }

    // ---- ASYNC: each thread stages its own 24B pred row into LDS (turns the
    // 6-float-strided reads into one coalesced async burst per wave).
    {
        unsigned long long gp = (unsigned long long)(pred + (size_t)i * 6);
        unsigned la = lds_addr_of(&spred[tid * 6]);
        asm volatile("global_load_async_to_lds_b64 %0, %1, off"
                     :: "v"(la), "v"(gp) : "memory");
        asm volatile("global_load_async_to_lds_b64 %0, %1, off offset:8"
                     :: "v"(la), "v"(gp) : "memory");
        asm volatile("global_load_async_to_lds_b64 %0, %1, off offset:16"
                     :: "v"(la), "v"(gp) : "memory");
    }

    asm volatile("s_wait_asynccnt 0x0" ::: "memory");
    __builtin_amdgcn_s_wait_tensorcnt(0);     // no-op for waves with cnt==0
    __syncthreads();

    const float ax1 = spred[tid * 6 + 0];
    const float ay1 = spred[tid * 6 + 1];
    const float ax2 = spred[tid * 6 + 2];
    const float ay2 = spred[tid * 6 + 3];
    const float areaA = (ax2 - ax1) * (ay2 - ay1);

    // ---- Scan the shared tile. Uniform j across the wave -> LDS broadcast.
    bool matched = false;
    for (int j0 = 0; j0 < M_BOX; j0 += 64) {
        #pragma unroll 4
        for (int j = j0; j < j0 + 64; ++j) {
            float4 b    = sbox[j];
            float ix1   = fmaxf(ax1, b.x);
            float iy1   = fmaxf(ay1, b.y);
            float ix2   = fminf(ax2, b.z);
            float iy2   = fminf(ay2, b.w);
            float inter = fmaxf(ix2 - ix1, 0.0f) * fmaxf(iy2 - iy1, 0.0f);
            float areaB = (b.z - b.x) * (b.w - b.y);
            float iou   = inter / (areaA + areaB - inter + EPS_F); // IEEE div, parity w/ ref
            matched = matched || (iou > IOU_THR);
        }
        // Wave-uniform early exit once every lane has a match (wave32 ballot).
        if (__builtin_amdgcn_ballot_w32(!matched) == 0u) break;
    }

    if (matched) atomicAdd(&counts[i], 1u);   // integer atomic: deterministic
}

__global__ void zero_counts_kernel(unsigned* __restrict__ counts) {
    int i = blockIdx.x * blockDim.x + threadIdx.x;
    if (i < N_PRED) counts[i] = 0u;
}

__global__ void finalize_kernel(const unsigned* __restrict__ counts,
                                float* __restrict__ out) {
    int i = blockIdx.x * blockDim.x + threadIdx.x;
    if (i < N_PRED) out[i] = (float)counts[i] * (1.0f / (float)T_RUNS);
}

extern "C" void kernel_launch(void* const* d_in, const int* in_sizes, int n_in,
                              void* d_out, int out_size, void* d_ws, size_t ws_size,
                              hipStream_t stream) {
    const float* pred          = (const float*)d_in[0]; // (2048, 6)
    const float* dropout_preds = (const float*)d_in[1]; // (16, 2048, 6)
    // d_in[2] = dropout_cls_confs — unused by the reference.
    float*    out    = (float*)d_out;                   // (2048)
    unsigned* counts = (unsigned*)d_ws;                 // 2048 * 4 B scratch

    zero_counts_kernel<<<(N_PRED + 255) / 256, 256, 0, stream>>>(counts);

    dim3 grid(TILES, T_RUNS);
    iou_match_kernel<<<grid, BLOCK, 0, stream>>>(pred, dropout_preds, counts);

    finalize_kernel<<<(N_PRED + 255) / 256, 256, 0, stream>>>(counts, out);
}